// DTWLoss_60206851555474
// MI455X (gfx1250) — compile-verified
//
#include <hip/hip_runtime.h>
#include <hip/hip_bf16.h>

typedef float v2f __attribute__((ext_vector_type(2)));
typedef float v8f __attribute__((ext_vector_type(8)));

#define BATCH 64
#define SEQN  256
#define SEQM  256
#define DIM   128

// Dynamic LDS layout (floats):
//   cost : [0, 65536)              256x256 cost matrix (256 KB)
//   x2   : [65536, 65792)          row norms of x
//   y2   : [65792, 66048)          row norms of y
//   diag : [66048, 66048 + 3*264)  three anti-diagonal buffers (257 used, padded)
#define LDS_FLOATS (65536 + 256 + 256 + 3 * 264)

__global__ __launch_bounds__(256)
void dtw_batch_kernel(const float* __restrict__ x,
                      const float* __restrict__ y,
                      float* __restrict__ partial) {
    extern __shared__ float lds[];
    float* cost = lds;                 // 256*256
    float* x2   = lds + 65536;         // 256
    float* y2   = x2 + 256;            // 256
    float* diag = y2 + 256;            // 3 * 264

    const int b    = blockIdx.x;
    const int t    = threadIdx.x;      // 0..255
    const int lane = t & 31;
    const int wave = t >> 5;           // 0..7

    const float* xb = x + (size_t)b * SEQN * DIM;
    const float* yb = y + (size_t)b * SEQM * DIM;

    // ---------- Phase 0: squared row norms (one row per thread) ----------
    {
        const float4* xr = (const float4*)(xb + t * DIM);
        const float4* yr = (const float4*)(yb + t * DIM);
        float sx = 0.f, sy = 0.f;
#pragma unroll 8
        for (int k = 0; k < DIM / 4; ++k) {
            float4 a = xr[k];
            float4 c = yr[k];
            sx += a.x * a.x + a.y * a.y + a.z * a.z + a.w * a.w;
            sy += c.x * c.x + c.y * c.y + c.z * c.z + c.w * c.w;
        }
        x2[t] = sx;
        y2[t] = sy;
    }
    __syncthreads();

    // ---------- Phase 1: cost = x2 + y2 - 2 * X @ Y^T via f32 WMMA ----------
    // 256 16x16 output tiles; 8 waves -> 32 tiles/wave; K=128 -> 32 WMMA/tile.
    {
        const int lo = lane & 15;      // M (for A) / N (for B,C,D)
        const int hi = lane >> 4;      // selects K pair / M+8 for C,D
        for (int tile = wave; tile < 256; tile += 8) {
            const int ti = tile >> 4;
            const int tj = tile & 15;
            // A fragment: 16x4 f32, lane L holds row M=L%16, VGPR k-pair by L/16
            const float* aptr = xb + (ti * 16 + lo) * DIM + 2 * hi;
            // B fragment: 4x16 f32 of Y^T -> column n of B is row n of Y
            const float* bptr = yb + (tj * 16 + lo) * DIM + 2 * hi;
            v8f acc = {};
#pragma unroll
            for (int kk = 0; kk < DIM / 4; ++kk) {
                v2f av = *(const v2f*)(aptr + kk * 4);
                v2f bv = *(const v2f*)(bptr + kk * 4);
                // 8 args: (neg_a, A, neg_b, B, c_mod, C, reuse_a, reuse_b)
                acc = __builtin_amdgcn_wmma_f32_16x16x4_f32(
                    false, av, false, bv, (short)0, acc, false, false);
            }
            const int j = tj * 16 + lo;
#pragma unroll
            for (int v = 0; v < 8; ++v) {
                const int i = ti * 16 + v + 8 * hi;   // C/D layout: M = v + 8*(lane/16)
                cost[i * 256 + j] = x2[i] + y2[j] - 2.0f * acc[v];
            }
        }
    }
    __syncthreads();

    // ---------- Phase 2: soft-DTW wavefront DP over anti-diagonals ----------
    const float INF = __builtin_inff();
    float* d0 = diag;          // diagonal d-2
    float* d1 = diag + 264;    // diagonal d-1
    float* d2 = diag + 528;    // diagonal d (being written)

    // diag(d)[j] == R[d-j][j].  Init d=0 and d=1: R[0][0]=0, borders = INF.
    d0[t + 1] = INF;
    d1[t + 1] = INF;
    if (t == 0) { d0[0] = 0.0f; d1[0] = INF; }
    __syncthreads();

    for (int d = 2; d <= SEQN + SEQM; ++d) {
        const int j = t + 1;           // 1..256
        const int i = d - j;
        float val = INF;
        if (i >= 1 && i <= SEQN) {
            const float up   = d1[j];       // R[i-1][j]
            const float left = d1[j - 1];   // R[i][j-1]
            const float dg   = d0[j - 1];   // R[i-1][j-1]
            const float m = fminf(up, fminf(left, dg));
            float sm;
            if (m < INF) {
                // gamma = 1: softmin = m - log(sum exp(m - r))
                sm = m - __logf(__expf(m - up) + __expf(m - left) + __expf(m - dg));
            } else {
                sm = INF;
            }
            val = cost[(i - 1) * 256 + (j - 1)] + sm;
        }
        d2[j] = val;
        if (t == 0) d2[0] = INF;            // R[d][0] = INF
        if (d == SEQN + SEQM && t == 255) partial[b] = val;  // R[256][256]
        __syncthreads();
        float* tmp = d0; d0 = d1; d1 = d2; d2 = tmp;
    }
}

__global__ void dtw_reduce_kernel(const float* __restrict__ partial,
                                  float* __restrict__ out) {
    if (blockIdx.x == 0 && threadIdx.x == 0) {
        float s = 0.0f;
        for (int i = 0; i < BATCH; ++i) s += partial[i];  // fixed order: deterministic
        out[0] = s * (1.0f / (float)BATCH);
    }
}

extern "C" void kernel_launch(void* const* d_in, const int* in_sizes, int n_in,
                              void* d_out, int out_size, void* d_ws, size_t ws_size,
                              hipStream_t stream) {
    const float* x = (const float*)d_in[0];   // (64, 256, 128) f32
    const float* y = (const float*)d_in[1];   // (64, 256, 128) f32
    float* partial = (float*)d_ws;            // 64 floats of scratch
    float* out     = (float*)d_out;           // scalar f32

    const size_t lds_bytes = (size_t)LDS_FLOATS * sizeof(float);  // ~267 KB < 320 KB/WGP
    (void)hipFuncSetAttribute((const void*)dtw_batch_kernel,
                              hipFuncAttributeMaxDynamicSharedMemorySize,
                              (int)lds_bytes);

    dtw_batch_kernel<<<BATCH, 256, lds_bytes, stream>>>(x, y, partial);
    dtw_reduce_kernel<<<1, 32, 0, stream>>>(partial, out);
}